// DualHopGCNModel_5858335391835
// MI455X (gfx1250) — compile-verified
//
#include <hip/hip_runtime.h>

typedef __attribute__((ext_vector_type(2))) float v2f;
typedef __attribute__((ext_vector_type(8))) float v8f;

#define HIDC 128

// ---------------------------------------------------------------------------
// GEMM: C[M,NC] = A[M,128] @ B[128,NC] (+ bias). One 16x16 tile per wave32,
// fully unrolled K-loop of V_WMMA_F32_16X16X4_F32 (32 WMMAs).
// Out-of-range rows/cols are handled by clamping load addresses (exact: a
// clamped A-row only affects that row's outputs, a clamped B-col only that
// lane's output column) and masking stores — no exec divergence in the loop.
// ---------------------------------------------------------------------------
template <int NC>
__global__ __launch_bounds__(256) void gemm_wmma_f32(
    const float* __restrict__ A, const float* __restrict__ B,
    const float* __restrict__ bias, float* __restrict__ C, int M)
{
    constexpr int K = HIDC;
    constexpr int tilesN = (NC + 15) >> 4;
    const int tilesM = (M + 15) >> 4;
    const int wid = blockIdx.x * 8 + (threadIdx.x >> 5);
    if (wid >= tilesM * tilesN) return;          // wave-uniform exit
    const int tm = wid / tilesN, tn = wid % tilesN;
    const int m0 = tm << 4, n0 = tn << 4;
    const int lane = threadIdx.x & 31;
    const int half = lane >> 4;                  // 0: K pair {0,1}, 1: {2,3}
    const int l16  = lane & 15;

    const int arow   = min(m0 + l16, M - 1);     // clamped load row
    const int bcol_r = n0 + l16;                 // real output column
    const int bcol   = min(bcol_r, NC - 1);      // clamped load column

    const float* Ap = A + (size_t)arow * K + 2 * half;   // 8B aligned (even k)
    const float* Bp = B + (size_t)(2 * half) * NC + bcol;

    v8f acc = {};
#pragma unroll
    for (int k0 = 0; k0 < K; k0 += 4) {
        v2f a = *(const v2f*)(Ap + k0);          // global_load_b64
        v2f b;
        b.x = Bp[k0 * NC];                       // immediate offsets (NC const)
        b.y = Bp[(k0 + 1) * NC];
        acc = __builtin_amdgcn_wmma_f32_16x16x4_f32(
                  false, a, false, b, (short)0, acc, false, false);
    }

    if (bcol_r < NC) {
        const float bv = bias ? bias[bcol_r] : 0.0f;
#pragma unroll
        for (int v = 0; v < 8; ++v) {
            const int r = m0 + v + 8 * half;     // C/D layout: VGPR v, lane half
            if (r < M) C[(size_t)r * NC + bcol_r] = acc[v] + bv;
        }
    }
}

// ---------------------------------------------------------------------------
// Helpers
// ---------------------------------------------------------------------------
__global__ void zero_f32(float* __restrict__ p, int n) {
    int i = blockIdx.x * blockDim.x + threadIdx.x;
    if (i < n) p[i] = 0.0f;
}

__global__ void degree_count(const int* __restrict__ row, int E,
                             float* __restrict__ deg) {
    int e = blockIdx.x * blockDim.x + threadIdx.x;
    if (e < E) atomicAdd(&deg[row[e]], 1.0f);
}

__global__ void invert_deg(float* __restrict__ d, int n) {
    int i = blockIdx.x * blockDim.x + threadIdx.x;
    if (i < n) d[i] = 1.0f / (d[i] + 1e-10f);
}

// agg[col[e]] += m[row[e]]  — thread handles 4 channels (float4 gather, L2 atomics)
__global__ __launch_bounds__(256) void scatter_add_edges(
    const float* __restrict__ m, const int* __restrict__ ei, int E,
    float* __restrict__ agg)
{
    int t  = blockIdx.x * 256 + threadIdx.x;
    int e  = t >> 5;
    int c4 = (t & 31) << 2;
    if (e >= E) return;
    int r  = ei[e];
    int cl = ei[E + e];
    const float4 mv = *(const float4*)(m + (size_t)r * HIDC + c4);
    float* dst = agg + (size_t)cl * HIDC + c4;
    atomicAdd(dst + 0, mv.x);
    atomicAdd(dst + 1, mv.y);
    atomicAdd(dst + 2, mv.z);
    atomicAdd(dst + 3, mv.w);
}

__global__ void bias_relu(float* __restrict__ agg, const float* __restrict__ b,
                          int total) {
    int i = blockIdx.x * blockDim.x + threadIdx.x;
    if (i < total) {
        float v = agg[i] + b[i & (HIDC - 1)];
        agg[i] = fmaxf(v, 0.0f);
    }
}

// per edge: s = ||agg[row]-agg[col]||^2 ; gsum[row] += s. One wave32 per edge,
// lane covers 4 channels (B128 loads), shuffle tree reduction.
__global__ __launch_bounds__(256) void gamma_edges(
    const float* __restrict__ agg, const int* __restrict__ ei, int E,
    float* __restrict__ gsum)
{
    int wid  = blockIdx.x * 8 + (threadIdx.x >> 5);
    int lane = threadIdx.x & 31;
    if (wid >= E) return;
    int r = ei[wid];
    int c = ei[E + wid];
    const float4 a = *(const float4*)(agg + (size_t)r * HIDC + lane * 4);
    const float4 b = *(const float4*)(agg + (size_t)c * HIDC + lane * 4);
    float dx = a.x - b.x, dy = a.y - b.y, dz = a.z - b.z, dw = a.w - b.w;
    float s = dx * dx + dy * dy + dz * dz + dw * dw;
#pragma unroll
    for (int off = 16; off > 0; off >>= 1)
        s += __shfl_down(s, off, 32);
    if (lane == 0) atomicAdd(&gsum[r], s);
}

__global__ void gamma_node(const float* __restrict__ gsum,
                           const float* __restrict__ inv,
                           float* __restrict__ g, int n) {
    int i = blockIdx.x * blockDim.x + threadIdx.x;
    if (i < n) g[i] = tanhf(gsum[i] * inv[i]);
}

__global__ void update_h(float* __restrict__ h, const float* __restrict__ agg,
                         const float* __restrict__ xs,
                         const float* __restrict__ g1,
                         const float* __restrict__ g2, int total) {
    int i = blockIdx.x * blockDim.x + threadIdx.x;
    if (i < total) {
        int node = i >> 7;                        // HIDC == 128
        float gs = g1[node], gq = g2[node];
        h[i] = (h[i] + gs * agg[i] + gq * xs[i]) / (1.0f + gs + gq);
    }
}

// ---------------------------------------------------------------------------
// Launch
// ---------------------------------------------------------------------------
extern "C" void kernel_launch(void* const* d_in, const int* in_sizes, int n_in,
                              void* d_out, int out_size, void* d_ws, size_t ws_size,
                              hipStream_t stream) {
    const float* x      = (const float*)d_in[0];
    const float* x0     = (const float*)d_in[1];
    const int*   ei     = (const int*)d_in[2];
    const int*   ei2    = (const int*)d_in[3];
    const float* in_W   = (const float*)d_in[4];
    const float* skip_W = (const float*)d_in[5];
    const float* conv_W = (const float*)d_in[6];
    const float* conv_b = (const float*)d_in[7];
    const float* fc_W   = (const float*)d_in[8];
    const float* fc_b   = (const float*)d_in[9];
    float* out = (float*)d_out;

    const int N  = in_sizes[0] / HIDC;   // 20000
    const int E1 = in_sizes[2] / 2;      // 120000
    const int E2 = in_sizes[3] / 2;      // data-dependent two-hop edge count

    const size_t nh = (size_t)N * HIDC;
    float* h     = (float*)d_ws;
    float* xskip = h + nh;
    float* agg   = xskip + nh;
    float* mbuf  = agg + nh;
    float* inv1  = mbuf + nh;
    float* inv2  = inv1 + N;
    float* gsum1 = inv2 + N;
    float* gsum2 = gsum1 + N;
    float* g1    = gsum2 + N;
    float* g2    = g1 + N;

    const dim3 blk(256);
    auto gemm_blocks = [](int M, int Nc) {
        int tiles = ((M + 15) / 16) * ((Nc + 15) / 16);
        return dim3((tiles + 7) / 8);
    };
    const int nhi = (int)nh;

    // h = x @ in_W ; h0 (mbuf) = x0 @ in_W ; x_skip = h0 @ skip_W
    gemm_wmma_f32<HIDC><<<gemm_blocks(N, HIDC), blk, 0, stream>>>(x,    in_W,   nullptr, h,     N);
    gemm_wmma_f32<HIDC><<<gemm_blocks(N, HIDC), blk, 0, stream>>>(x0,   in_W,   nullptr, mbuf,  N);
    gemm_wmma_f32<HIDC><<<gemm_blocks(N, HIDC), blk, 0, stream>>>(mbuf, skip_W, nullptr, xskip, N);

    // inverse degrees (accumulate in inv*, then invert in place)
    zero_f32<<<(N + 255) / 256, blk, 0, stream>>>(inv1, N);
    zero_f32<<<(N + 255) / 256, blk, 0, stream>>>(inv2, N);
    degree_count<<<(E1 + 255) / 256, blk, 0, stream>>>(ei,  E1, inv1);
    degree_count<<<(E2 + 255) / 256, blk, 0, stream>>>(ei2, E2, inv2);
    invert_deg<<<(N + 255) / 256, blk, 0, stream>>>(inv1, N);
    invert_deg<<<(N + 255) / 256, blk, 0, stream>>>(inv2, N);

    for (int l = 0; l < 2; ++l) {
        // m = h @ conv_W[l]
        gemm_wmma_f32<HIDC><<<gemm_blocks(N, HIDC), blk, 0, stream>>>(
            h, conv_W + (size_t)l * HIDC * HIDC, nullptr, mbuf, N);
        // x_agg = relu(scatter_add(m[row] -> col) + conv_b[l])
        zero_f32<<<(nhi + 255) / 256, blk, 0, stream>>>(agg, nhi);
        {
            long long threads = (long long)E1 * 32;
            scatter_add_edges<<<(unsigned)((threads + 255) / 256), blk, 0, stream>>>(mbuf, ei, E1, agg);
        }
        bias_relu<<<(nhi + 255) / 256, blk, 0, stream>>>(agg, conv_b + l * HIDC, nhi);
        // gamma terms for 1-hop and 2-hop edge sets
        zero_f32<<<(N + 255) / 256, blk, 0, stream>>>(gsum1, N);
        zero_f32<<<(N + 255) / 256, blk, 0, stream>>>(gsum2, N);
        gamma_edges<<<(E1 + 7) / 8, blk, 0, stream>>>(agg, ei,  E1, gsum1);
        gamma_edges<<<(E2 + 7) / 8, blk, 0, stream>>>(agg, ei2, E2, gsum2);
        gamma_node<<<(N + 255) / 256, blk, 0, stream>>>(gsum1, inv1, g1, N);
        gamma_node<<<(N + 255) / 256, blk, 0, stream>>>(gsum2, inv2, g2, N);
        // h = (h + gs*agg + gq*x_skip) / (1 + gs + gq)
        update_h<<<(nhi + 255) / 256, blk, 0, stream>>>(h, agg, xskip, g1, g2, nhi);
    }

    // out = h @ fc_W + fc_b   (NC = 40, partial tile handled by clamp+mask)
    gemm_wmma_f32<40><<<gemm_blocks(N, 40), blk, 0, stream>>>(h, fc_W, fc_b, out, N);
}